// RingConeChain_23691039605492
// MI455X (gfx1250) — compile-verified
//
#include <hip/hip_runtime.h>

#define NN 50000
#define DIM 128
#define NE 800000
#define NPOL 9
#define NLAYERS 3
#define LNEPS 1e-5f
#define DD (DIM * DIM)
#define PDD (NPOL * DIM * DIM)

typedef __attribute__((ext_vector_type(2))) float v2f;
typedef __attribute__((ext_vector_type(8))) float v8f;

// ---------------------------------------------------------------- elementwise
__global__ void k_zero_f(float* __restrict__ p, int n) {
  int i = blockIdx.x * blockDim.x + threadIdx.x;
  if (i < n) p[i] = 0.0f;
}
__global__ void k_zero_i(int* __restrict__ p, int n) {
  int i = blockIdx.x * blockDim.x + threadIdx.x;
  if (i < n) p[i] = 0;
}
__global__ void k_copy_f(float* __restrict__ d, const float* __restrict__ s, int n) {
  int i = blockIdx.x * blockDim.x + threadIdx.x;
  if (i < n) d[i] = s[i];
}
__global__ void k_final_add(const float* __restrict__ a, const float* __restrict__ b,
                            float* __restrict__ o, int n) {
  int i = blockIdx.x * blockDim.x + threadIdx.x;
  if (i < n) o[i] = a[i] + b[i];
}

// ---------------------------------------------------------------- degree
__global__ void k_deg_count(const int* __restrict__ src, float* __restrict__ deg, int e) {
  int i = blockIdx.x * blockDim.x + threadIdx.x;
  if (i < e) atomicAdd(&deg[src[i]], 1.0f);
}
__global__ void k_deg_norm(float* __restrict__ deg, int n) {
  int i = blockIdx.x * blockDim.x + threadIdx.x;
  if (i < n) deg[i] = 1.0f / fmaxf(deg[i], 1.0f);
}

// ---------------------------------------------------------------- polarity bucket sort
__global__ void k_pol_count(const int* __restrict__ rp, int* __restrict__ cnt, int n) {
  int i = blockIdx.x * blockDim.x + threadIdx.x;
  if (i < n) {
    int p = rp[i] % NPOL; if (p < 0) p += NPOL;
    atomicAdd(&cnt[p], 1);
  }
}
__global__ void k_scan(const int* __restrict__ cnt, int* __restrict__ bstart) {
  if (blockIdx.x == 0 && threadIdx.x == 0) {
    int s = 0;
    for (int p = 0; p < NPOL; ++p) { bstart[p] = s; s += cnt[p]; }
    bstart[NPOL] = s;
  }
}
__global__ void k_pol_fill(const int* __restrict__ rp, const int* __restrict__ bstart,
                           int* __restrict__ cursor, int* __restrict__ sorted, int n) {
  int i = blockIdx.x * blockDim.x + threadIdx.x;
  if (i < n) {
    int p = rp[i] % NPOL; if (p < 0) p += NPOL;
    int pos = atomicAdd(&cursor[p], 1);
    sorted[bstart[p] + pos] = i;
  }
}

// ---------------------------------------------------------------- WMMA f32 GEMMs
// V_WMMA_F32_16X16X4_F32 layouts (wave32):
//  A frag (16x4): lane l holds A[l&15][k + 2*(l>>4) + {0,1}]  (contiguous float2)
//  B frag (4x16): lane l holds B[k + 2*(l>>4) + {0,1}][l&15]
//  C tile: VGPR v, lanes 0-15 -> M=v, N=lane; lanes 16-31 -> M=v+8, N=lane-16

// Cmb[p] = (S[p]+dS[p]) @ Wr^T ; one 16x128 row band per block, 8 waves = 8 col tiles
__global__ __launch_bounds__(256) void k_combine(const float* __restrict__ S,
                                                 const float* __restrict__ dS,
                                                 const float* __restrict__ Wr,
                                                 float* __restrict__ Cmb) {
  const int p = blockIdx.y;
  const int t = threadIdx.x, lane = t & 31, w = t >> 5;
  const int half = lane >> 4, m = lane & 15, n0 = w * 16;
  const int row0 = blockIdx.x * 16;
  const float* __restrict__ A1 = S + (size_t)p * DD + (size_t)(row0 + m) * DIM;
  const float* __restrict__ A2 = dS + (size_t)p * DD + (size_t)(row0 + m) * DIM;
  const float* __restrict__ Bw = Wr + (size_t)(n0 + m) * DIM;  // B[j][n] = Wr[n][j]
  v8f acc = {};
#pragma unroll 4
  for (int k0 = 0; k0 < DIM; k0 += 4) {
    const int kk = k0 + 2 * half;
    v2f a = *(const v2f*)(A1 + kk);
    v2f a2 = *(const v2f*)(A2 + kk);
    a.x += a2.x; a.y += a2.y;
    v2f b = *(const v2f*)(Bw + kk);
    acc = __builtin_amdgcn_wmma_f32_16x16x4_f32(false, a, false, b, (short)0, acc, false, false);
  }
#pragma unroll
  for (int v = 0; v < 8; ++v)
    Cmb[(size_t)p * DD + (size_t)(row0 + v + 8 * half) * DIM + n0 + m] = acc[v];
}

// Y[row] = X[row] @ B[pol(row)] for rows gathered from polarity buckets.
// 64 rows per block (4 WMMA row sub-tiles per wave share each B fragment -> ~1.25 loads/WMMA).
// ADD_B2: B = B1 + B2 (R + dR). SCALE: multiply output row by scale[row] (deg_norm).
template <bool ADD_B2, bool SCALE>
__global__ __launch_bounds__(256) void k_pol_gemm(const float* __restrict__ X,
                                                  const float* __restrict__ B1,
                                                  const float* __restrict__ B2,
                                                  const float* __restrict__ scale,
                                                  const int* __restrict__ sorted,
                                                  const int* __restrict__ bstart,
                                                  float* __restrict__ Y) {
  const int p = blockIdx.y;
  __shared__ int s_rows[64];
  __shared__ int s_cnt;
  const int t = threadIdx.x;
  if (t == 0) s_cnt = bstart[p + 1] - bstart[p];
  __syncthreads();
  const int cnt = s_cnt;
  const int tile0 = blockIdx.x * 64;
  if (tile0 >= cnt) return;  // block-uniform
  if (t < 64) {
    int r = tile0 + t; if (r > cnt - 1) r = cnt - 1;  // clamp partial tile
    s_rows[t] = sorted[bstart[p] + r];
  }
  __syncthreads();
  const int lane = t & 31, w = t >> 5;
  const int half = lane >> 4, m = lane & 15, n0 = w * 16;
  const float* __restrict__ Arow[4];
#pragma unroll
  for (int mt = 0; mt < 4; ++mt) Arow[mt] = X + (size_t)s_rows[mt * 16 + m] * DIM;
  const float* __restrict__ Bp1 = B1 + (size_t)p * DD;
  const float* __restrict__ Bp2 = ADD_B2 ? (B2 + (size_t)p * DD) : nullptr;
  v8f acc[4] = {};
#pragma unroll 4
  for (int k0 = 0; k0 < DIM; k0 += 4) {
    const int kk = k0 + 2 * half;
    v2f b;
    b.x = Bp1[(size_t)kk * DIM + n0 + m];
    b.y = Bp1[(size_t)(kk + 1) * DIM + n0 + m];
    if (ADD_B2) {
      b.x += Bp2[(size_t)kk * DIM + n0 + m];
      b.y += Bp2[(size_t)(kk + 1) * DIM + n0 + m];
    }
#pragma unroll
    for (int mt = 0; mt < 4; ++mt) {
      v2f a = *(const v2f*)(Arow[mt] + kk);
      acc[mt] =
          __builtin_amdgcn_wmma_f32_16x16x4_f32(false, a, false, b, (short)0, acc[mt], false, false);
    }
  }
#pragma unroll
  for (int mt = 0; mt < 4; ++mt) {
#pragma unroll
    for (int v = 0; v < 8; ++v) {
      const int r = mt * 16 + v + 8 * half;
      if (tile0 + r < cnt) {
        const int grow = s_rows[r];
        float val = acc[mt][v];
        if (SCALE) val *= scale[grow];
        Y[(size_t)grow * DIM + n0 + m] = val;
      }
    }
  }
}

// Y = X @ W^T + bias (shared 128x128 W, row-major [out][in]); 64 rows per block.
__global__ __launch_bounds__(256) void k_gemm_wt_bias(const float* __restrict__ X,
                                                      const float* __restrict__ W,
                                                      const float* __restrict__ bias,
                                                      float* __restrict__ Y, int nrows) {
  const int tile0 = blockIdx.x * 64;
  const int t = threadIdx.x, lane = t & 31, w = t >> 5;
  const int half = lane >> 4, m = lane & 15, n0 = w * 16;
  const float* __restrict__ Arow[4];
#pragma unroll
  for (int mt = 0; mt < 4; ++mt) {
    int r = tile0 + mt * 16 + m; if (r > nrows - 1) r = nrows - 1;
    Arow[mt] = X + (size_t)r * DIM;
  }
  const float* __restrict__ Brow = W + (size_t)(n0 + m) * DIM;  // B[j][n] = W[n][j]
  v8f acc[4] = {};
#pragma unroll 4
  for (int k0 = 0; k0 < DIM; k0 += 4) {
    const int kk = k0 + 2 * half;
    v2f b = *(const v2f*)(Brow + kk);
#pragma unroll
    for (int mt = 0; mt < 4; ++mt) {
      v2f a = *(const v2f*)(Arow[mt] + kk);
      acc[mt] =
          __builtin_amdgcn_wmma_f32_16x16x4_f32(false, a, false, b, (short)0, acc[mt], false, false);
    }
  }
  const float bv = bias[n0 + m];
#pragma unroll
  for (int mt = 0; mt < 4; ++mt) {
#pragma unroll
    for (int v = 0; v < 8; ++v) {
      const int r = tile0 + mt * 16 + v + 8 * half;
      if (r < nrows) Y[(size_t)r * DIM + n0 + m] = acc[mt][v] + bv;
    }
  }
}

// ---------------------------------------------------------------- edge scatter
__global__ void k_scatter(const int* __restrict__ src, const int* __restrict__ tgt,
                          const float* __restrict__ Yv, float* __restrict__ agg, int e) {
  long long i = (long long)blockIdx.x * blockDim.x + threadIdx.x;
  if (i >= (long long)e * 32) return;
  const int ed = (int)(i >> 5);
  const int q = ((int)i & 31) * 4;
  const float4 v = *(const float4*)(Yv + (size_t)src[ed] * DIM + q);
  float* dst = agg + (size_t)tgt[ed] * DIM + q;
  atomicAdd(dst + 0, v.x);
  atomicAdd(dst + 1, v.y);
  atomicAdd(dst + 2, v.z);
  atomicAdd(dst + 3, v.w);
}

// ---------------------------------------------------------------- LayerNorm (wave32 per row)
template <bool RELU>
__global__ __launch_bounds__(256) void k_layernorm(float* __restrict__ Xio,
                                                   const float* __restrict__ g,
                                                   const float* __restrict__ b, int nrows) {
  const int row = blockIdx.x * 8 + (threadIdx.x >> 5);
  if (row >= nrows) return;
  const int lane = threadIdx.x & 31;
  float4 v = *(const float4*)(Xio + (size_t)row * DIM + lane * 4);
  float s = v.x + v.y + v.z + v.w;
  float s2 = v.x * v.x + v.y * v.y + v.z * v.z + v.w * v.w;
  for (int off = 16; off > 0; off >>= 1) {
    s += __shfl_xor(s, off, 32);
    s2 += __shfl_xor(s2, off, 32);
  }
  const float mean = s * (1.0f / DIM);
  const float inv = rsqrtf(s2 * (1.0f / DIM) - mean * mean + LNEPS);
  const float4 gg = *(const float4*)(g + lane * 4);
  const float4 bb = *(const float4*)(b + lane * 4);
  float4 o;
  o.x = (v.x - mean) * inv * gg.x + bb.x;
  o.y = (v.y - mean) * inv * gg.y + bb.y;
  o.z = (v.z - mean) * inv * gg.z + bb.z;
  o.w = (v.w - mean) * inv * gg.w + bb.w;
  if (RELU) {
    o.x = fmaxf(o.x, 0.0f); o.y = fmaxf(o.y, 0.0f);
    o.z = fmaxf(o.z, 0.0f); o.w = fmaxf(o.w, 0.0f);
  }
  *(float4*)(Xio + (size_t)row * DIM + lane * 4) = o;
}

// x = LN(res_scale*U + x) * g + b  (in place on xcur)
__global__ __launch_bounds__(256) void k_resid_ln(const float* __restrict__ U,
                                                  float* __restrict__ Xcur,
                                                  const float* __restrict__ res_scale, int l,
                                                  const float* __restrict__ g,
                                                  const float* __restrict__ b, int nrows) {
  const int row = blockIdx.x * 8 + (threadIdx.x >> 5);
  if (row >= nrows) return;
  const int lane = threadIdx.x & 31;
  const float rs = res_scale[l];
  const float4 u = *(const float4*)(U + (size_t)row * DIM + lane * 4);
  float4 x = *(const float4*)(Xcur + (size_t)row * DIM + lane * 4);
  float4 v;
  v.x = rs * u.x + x.x; v.y = rs * u.y + x.y;
  v.z = rs * u.z + x.z; v.w = rs * u.w + x.w;
  float s = v.x + v.y + v.z + v.w;
  float s2 = v.x * v.x + v.y * v.y + v.z * v.z + v.w * v.w;
  for (int off = 16; off > 0; off >>= 1) {
    s += __shfl_xor(s, off, 32);
    s2 += __shfl_xor(s2, off, 32);
  }
  const float mean = s * (1.0f / DIM);
  const float inv = rsqrtf(s2 * (1.0f / DIM) - mean * mean + LNEPS);
  const float4 gg = *(const float4*)(g + lane * 4);
  const float4 bb = *(const float4*)(b + lane * 4);
  float4 o;
  o.x = (v.x - mean) * inv * gg.x + bb.x;
  o.y = (v.y - mean) * inv * gg.y + bb.y;
  o.z = (v.z - mean) * inv * gg.z + bb.z;
  o.w = (v.w - mean) * inv * gg.w + bb.w;
  *(float4*)(Xcur + (size_t)row * DIM + lane * 4) = o;
}

// ---------------------------------------------------------------- host launcher
extern "C" void kernel_launch(void* const* d_in, const int* in_sizes, int n_in,
                              void* d_out, int out_size, void* d_ws, size_t ws_size,
                              hipStream_t stream) {
  const float* x0 = (const float*)d_in[0];
  const int* ei = (const int*)d_in[1];
  const int* src = ei;
  const int* tgt = ei + NE;
  const int* rp = (const int*)d_in[2];
  const float* Wr = (const float*)d_in[3];
  const float* S = (const float*)d_in[4];
  const float* dS = (const float*)d_in[5];
  const float* R = (const float*)d_in[6];
  const float* dR = (const float*)d_in[7];
  const float* res_scale = (const float*)d_in[8];
  const float* W1 = (const float*)d_in[9];
  const float* b1 = (const float*)d_in[10];
  const float* ln_g = (const float*)d_in[11];
  const float* ln_b = (const float*)d_in[12];
  const float* W2 = (const float*)d_in[13];
  const float* b2 = (const float*)d_in[14];
  const float* norm_g = (const float*)d_in[15];
  const float* norm_b = (const float*)d_in[16];
  float* out = (float*)d_out;

  char* ws = (char*)d_ws;
  size_t off = 0;
  auto carve = [&](size_t bytes) -> void* {
    void* p = ws + off;
    off += (bytes + 255) & ~(size_t)255;
    return p;
  };
  const size_t ND = (size_t)NN * DIM;
  float* xcur = (float*)carve(ND * 4);
  float* ybuf = (float*)carve(ND * 4);
  float* aggbuf = (float*)carve(ND * 4);
  float* tbuf = (float*)carve(ND * 4);
  float* degnorm = (float*)carve(NN * 4);
  float* Cmb = (float*)carve((size_t)PDD * 4);
  int* sorted = (int*)carve(NN * 4);
  int* cnt = (int*)carve(NPOL * 4);
  int* bstart = (int*)carve((NPOL + 1) * 4);
  int* cursor = (int*)carve(NPOL * 4);

  const int T = 256;
  const int gND = (int)((ND + T - 1) / T);
  const int gN = (NN + T - 1) / T;
  const int gE = (NE + T - 1) / T;
  const int nTiles64 = (NN + 63) / 64;
  const int gLN = (NN + 7) / 8;
  const long long scatWork = (long long)NE * 32;
  const int gScat = (int)((scatWork + T - 1) / T);

  // --- degree + polarity buckets (re-done every call; deterministic) ---
  k_zero_f<<<gN, T, 0, stream>>>(degnorm, NN);
  k_zero_i<<<1, T, 0, stream>>>(cnt, NPOL);
  k_zero_i<<<1, T, 0, stream>>>(cursor, NPOL);
  k_deg_count<<<gE, T, 0, stream>>>(src, degnorm, NE);
  k_deg_norm<<<gN, T, 0, stream>>>(degnorm, NN);
  k_pol_count<<<gN, T, 0, stream>>>(rp, cnt, NN);
  k_scan<<<1, 1, 0, stream>>>(cnt, bstart);
  k_pol_fill<<<gN, T, 0, stream>>>(rp, bstart, cursor, sorted, NN);
  k_copy_f<<<gND, T, 0, stream>>>(xcur, x0, (int)ND);

  for (int l = 0; l < NLAYERS; ++l) {
    // Cmb[p] = (S+dS)[l,p] @ Wr[l]^T
    k_combine<<<dim3(DIM / 16, NPOL), T, 0, stream>>>(S + (size_t)l * PDD, dS + (size_t)l * PDD,
                                                      Wr + (size_t)l * DD, Cmb);
    // y = x @ Cmb[pol]
    k_pol_gemm<false, false><<<dim3(nTiles64, NPOL), T, 0, stream>>>(xcur, Cmb, nullptr, nullptr,
                                                                     sorted, bstart, ybuf);
    // agg[tgt] += y[src]
    k_zero_f<<<gND, T, 0, stream>>>(aggbuf, (int)ND);
    k_scatter<<<gScat, T, 0, stream>>>(src, tgt, ybuf, aggbuf, NE);
    // out = (agg @ (R+dR)[pol]) * deg_norm   -> ybuf
    k_pol_gemm<true, true><<<dim3(nTiles64, NPOL), T, 0, stream>>>(
        aggbuf, R + (size_t)l * PDD, dR + (size_t)l * PDD, degnorm, sorted, bstart, ybuf);
    // MLP: t = out @ W1^T + b1 ; h = relu(LN(t)) ; u = h @ W2^T + b2
    k_gemm_wt_bias<<<nTiles64, T, 0, stream>>>(ybuf, W1 + (size_t)l * DD, b1 + (size_t)l * DIM,
                                               tbuf, NN);
    k_layernorm<true><<<gLN, T, 0, stream>>>(tbuf, ln_g + (size_t)l * DIM,
                                             ln_b + (size_t)l * DIM, NN);
    k_gemm_wt_bias<<<nTiles64, T, 0, stream>>>(tbuf, W2 + (size_t)l * DD, b2 + (size_t)l * DIM,
                                               aggbuf, NN);
    // x = LN(res*u + x)
    k_resid_ln<<<gLN, T, 0, stream>>>(aggbuf, xcur, res_scale, l, norm_g + (size_t)l * DIM,
                                      norm_b + (size_t)l * DIM, NN);
  }
  k_final_add<<<gND, T, 0, stream>>>(xcur, x0, out, (int)ND);
}